// TemporalAttention_80247168958702
// MI455X (gfx1250) — compile-verified
//
#include <hip/hip_runtime.h>

// TemporalAttention fused pipeline for MI455X (gfx1250, wave32, WMMA + TDM).
//   Kernel 1: Q/K/V = x @ W{q,k,v} + b   via V_WMMA_F32_16X16X4_F32
//   Kernel 2: flash-attention with TDM double-buffered K staging
//             (tensor_load_to_lds, TENSORcnt) + residual + LayerNorm
// fp32 throughout (reference is fp32; LN epilogue punishes low precision).

#define D_DIM 512
#define S_LEN 2048
#define BATCH 8
// Effective LDS row stride 516 floats (516 % 64 banks = 4 -> conflict-free
// transposed reads). K rows arrive via TDM with pad_interval=256dw,
// pad_amount=2dw: [0..255] pad2 [256..511] pad2  => same 516-float stride,
// element kk lives at kk + 2*(kk>>8).
#define DPAD  516
#define KBUFF (32 * DPAD)

typedef __attribute__((ext_vector_type(2))) float v2f;
typedef __attribute__((ext_vector_type(8))) float v8f;

__device__ __forceinline__ v8f wmma_f32_k4(v2f a, v2f b, v8f c) {
  return __builtin_amdgcn_wmma_f32_16x16x4_f32(
      false, a, false, b, (short)0, c, false, false);
}

#if __has_builtin(__builtin_amdgcn_tensor_load_to_lds)
#define HAVE_TDM 1
typedef __attribute__((ext_vector_type(4))) unsigned int tdm_u32x4;
typedef __attribute__((ext_vector_type(8))) int tdm_i32x8;
typedef __attribute__((ext_vector_type(4))) int tdm_i32x4;

// DMA a [32 x 512] fp32 tile (rows of K) into LDS with mid-row padding.
__device__ __forceinline__ void tdm_load_k_block(const float* gsrc, float* lds_dst) {
  const unsigned long long ga = (unsigned long long)(uintptr_t)gsrc;
  const unsigned int ldsoff = (unsigned int)(uintptr_t)lds_dst; // flat[31:0] == LDS offset
  tdm_u32x4 g0;
  g0[0] = 1u;                                        // count=1 (valid user descriptor)
  g0[1] = ldsoff;                                    // lds_addr (bytes)
  g0[2] = (unsigned int)(ga & 0xFFFFFFFFu);          // global_addr[31:0]
  g0[3] = (unsigned int)((ga >> 32) & 0x01FFFFFFu)   // global_addr[56:32]
          | (2u << 30);                              // type = 2 ("image")
  tdm_i32x8 g1;
  g1[0] = (int)((2u << 16)      // data_size = 4 bytes
              | (1u << 20)      // pad_enable
              | (7u << 22)      // pad_interval: 256 DWORDs
              | (1u << 25));    // pad_amount: 2 DWORDs
  g1[1] = (int)(512u << 16);    // tensor_dim0 = 512 (bits[79:48], low half)
  g1[2] = (int)(2048u << 16);   // tensor_dim0 hi=0 | tensor_dim1 = 2048 (low half)
  g1[3] = (int)(512u << 16);    // tensor_dim1 hi=0 | tile_dim0 = 512
  g1[4] = 32;                   // tile_dim1 = 32, tile_dim2 = 0
  g1[5] = 512;                  // tensor_dim0_stride = 512 (low 32 of 48)
  g1[6] = 0;                    // stride0 hi | tensor_dim1_stride lo (unused, 2D)
  g1[7] = 0;
  const tdm_i32x4 z4 = {0, 0, 0, 0};
  const tdm_i32x8 z8 = {0, 0, 0, 0, 0, 0, 0, 0};
  // 6-arg form (clang-23 / therock-10.0 headers): groups 2,3,(4) + cpol
  __builtin_amdgcn_tensor_load_to_lds(g0, g1, z4, z4, z8, 0);
}
#else
#define HAVE_TDM 0
#endif

// ---------------------------------------------------------------------------
// Kernel 1: fused QKV projection. One wave computes a 16x64 strip of one of
// Q/K/V. 24576 waves total (1024 m-tiles x 8 n-strips x 3 projections).
// ---------------------------------------------------------------------------
__global__ void __launch_bounds__(256)
qkv_proj_kernel(const float* __restrict__ x,
                const float* __restrict__ Wq, const float* __restrict__ bq,
                const float* __restrict__ Wk, const float* __restrict__ bk,
                const float* __restrict__ Wv, const float* __restrict__ bv,
                float* __restrict__ Qo, float* __restrict__ Ko, float* __restrict__ Vo)
{
  const int lane = threadIdx.x & 31;
  const int gw   = blockIdx.x * 8 + (threadIdx.x >> 5);
  const int nstrip = gw & 7;
  const int mtile  = (gw >> 3) & 1023;
  const int p      = gw >> 13;

  const float* W    = (p == 0) ? Wq : (p == 1) ? Wk : Wv;
  const float* bias = (p == 0) ? bq : (p == 1) ? bk : bv;
  float*       out  = (p == 0) ? Qo : (p == 1) ? Ko : Vo;

  const int m  = lane & 15;
  const int hi = lane >> 4;
  const int row0 = mtile * 16;
  const int col0 = nstrip * 64;

  v8f acc0 = {}, acc1 = {}, acc2 = {}, acc3 = {};
  const float* ap = x + (size_t)(row0 + m) * D_DIM + 2 * hi;
  const float* bp = W + (size_t)(2 * hi) * D_DIM + col0 + m;

  for (int k = 0; k < D_DIM; k += 4) {
    v2f a = *(const v2f*)(ap + k);
    const float* bk0 = bp + (size_t)k * D_DIM;
    v2f b0; b0.x = bk0[0];  b0.y = bk0[D_DIM + 0];
    v2f b1; b1.x = bk0[16]; b1.y = bk0[D_DIM + 16];
    v2f b2; b2.x = bk0[32]; b2.y = bk0[D_DIM + 32];
    v2f b3; b3.x = bk0[48]; b3.y = bk0[D_DIM + 48];
    acc0 = wmma_f32_k4(a, b0, acc0);
    acc1 = wmma_f32_k4(a, b1, acc1);
    acc2 = wmma_f32_k4(a, b2, acc2);
    acc3 = wmma_f32_k4(a, b3, acc3);
  }

  v8f accs[4] = {acc0, acc1, acc2, acc3};
  float* op = out + (size_t)row0 * D_DIM + col0;
  #pragma unroll
  for (int nt = 0; nt < 4; ++nt) {
    const int n = nt * 16 + m;
    const float bb = bias[col0 + n];
    #pragma unroll
    for (int v = 0; v < 8; ++v)
      op[(size_t)(v + 8 * hi) * D_DIM + n] = accs[nt][v] + bb;
  }
}

// ---------------------------------------------------------------------------
// Kernel 2: flash attention + residual + LayerNorm.
// One workgroup (8 waves) per (batch, 64-query block). 273KB of the 320KB WGP
// LDS: Q tile + double-buffered TDM-loaded K blocks + score tile.
// ---------------------------------------------------------------------------
__global__ void __launch_bounds__(256)
flash_attn_ln_kernel(const float* __restrict__ Qg, const float* __restrict__ Kg,
                     const float* __restrict__ Vg, const float* __restrict__ x,
                     const int* __restrict__ mask,
                     const float* __restrict__ gamma, const float* __restrict__ beta,
                     float* __restrict__ out)
{
  extern __shared__ float smem[];
  float* Qs     = smem;                    // 64 x DPAD (reused as residual buffer)
  float* KsBuf  = smem + 64 * DPAD;        // 2 x (32 x DPAD), double buffered
  float* Ps     = KsBuf + 2 * KBUFF;       // 64 x 32 scores / probabilities
  float* mstat  = Ps + 64 * 32;            // 64
  float* lstat  = mstat + 64;              // 64
  float* alphas = lstat + 64;              // 64

  const int tid  = threadIdx.x;
  const int lane = tid & 31;
  const int wave = tid >> 5;
  const int m  = lane & 15;
  const int hi = lane >> 4;
  const int b  = blockIdx.y;
  const int q0 = blockIdx.x * 64;

  const float* Kbase = Kg + (size_t)b * S_LEN * D_DIM;

#if HAVE_TDM
  if (wave == 0) tdm_load_k_block(Kbase, KsBuf);     // async preload of K block 0
#endif

  // Stage Q tile (coalesced)
  const float* Qrow = Qg + ((size_t)b * S_LEN + q0) * D_DIM;
  for (int idx = tid; idx < 64 * D_DIM; idx += 256) {
    const int r = idx >> 9, c = idx & (D_DIM - 1);
    Qs[r * DPAD + c] = Qrow[(size_t)r * D_DIM + c];
  }
  if (tid < 64) { mstat[tid] = -3.0e38f; lstat[tid] = 0.0f; }

  v8f o[16];
  #pragma unroll
  for (int i = 0; i < 16; ++i) { v8f z = {}; o[i] = z; }
  const int ocol0 = wave * 64;

  const int smi = wave >> 1;
  const int snt = wave & 1;

  __syncthreads();

  int it = 0;
  for (int kb = 0; kb < S_LEN; kb += 32, ++it) {
    float* Ksc = KsBuf + (it & 1) * KBUFF;

#if HAVE_TDM
    if (wave == 0) {
      if (kb + 32 < S_LEN) {
        // Kick off DMA of next K block into the other buffer, then wait for
        // the current buffer (in-order TENSORcnt: <=1 outstanding => done).
        tdm_load_k_block(Kbase + (size_t)(kb + 32) * D_DIM,
                         KsBuf + ((it + 1) & 1) * KBUFF);
        __builtin_amdgcn_s_wait_tensorcnt(1);
      } else {
        __builtin_amdgcn_s_wait_tensorcnt(0);
      }
    }
#else
    for (int idx = tid; idx < 32 * D_DIM; idx += 256) {
      const int r = idx >> 9, c = idx & (D_DIM - 1);
      Ksc[r * DPAD + c + 2 * (c >> 8)] = Kbase[(size_t)(kb + r) * D_DIM + c];
    }
#endif
    // Pull next V block toward the caches while we compute (global_prefetch_b8)
    if (kb + 32 < S_LEN)
      __builtin_prefetch((const char*)(Vg + ((size_t)b * S_LEN + kb + 32) * D_DIM)
                             + (size_t)tid * 256, 0, 1);
    __syncthreads();

    // Scores: S[16x16] = Q[16x512] * K^T[512x16], one tile per wave.
    // K element kk of row n sits at n*DPAD + kk + 2*(kk>>8).
    v8f s = {};
    const float* qa  = Qs  + (smi * 16 + m) * DPAD + 2 * hi;
    const float* kbp = Ksc + (snt * 16 + m) * DPAD + 2 * hi;
    for (int k = 0; k < 256; k += 4) {
      v2f a  = *(const v2f*)(qa + k);
      v2f bb = *(const v2f*)(kbp + k);
      s = wmma_f32_k4(a, bb, s);
    }
    for (int k = 256; k < 512; k += 4) {
      v2f a  = *(const v2f*)(qa + k);
      v2f bb = *(const v2f*)(kbp + k + 2);
      s = wmma_f32_k4(a, bb, s);
    }
    const int key = kb + snt * 16 + m;
    const float madd = (mask[b * S_LEN + key] == 0) ? -1.0e9f : 0.0f;
    #pragma unroll
    for (int v = 0; v < 8; ++v)
      Ps[(smi * 16 + v + 8 * hi) * 32 + snt * 16 + m] = s[v] * 0.04419417382415922f + madd;
    __syncthreads();

    // Online softmax: one thread per query row
    if (tid < 64) {
      float* pr = Ps + tid * 32;
      float mb = -3.0e38f;
      for (int j = 0; j < 32; ++j) mb = fmaxf(mb, pr[j]);
      const float mo = mstat[tid];
      const float mn = fmaxf(mo, mb);
      const float al = __expf(mo - mn);
      float sum = 0.0f;
      for (int j = 0; j < 32; ++j) { const float pv = __expf(pr[j] - mn); pr[j] = pv; sum += pv; }
      mstat[tid]  = mn;
      lstat[tid]  = lstat[tid] * al + sum;
      alphas[tid] = al;
    }
    __syncthreads();

    // Rescale O and accumulate O += P[64x32] * V[32x64-slice]
    const float* vbase = Vg + ((size_t)b * S_LEN + kb) * D_DIM + ocol0 + m;
    #pragma unroll
    for (int mi = 0; mi < 4; ++mi) {
      float arow[8];
      #pragma unroll
      for (int v = 0; v < 8; ++v) arow[v] = alphas[mi * 16 + v + 8 * hi];
      const float* pa = Ps + (mi * 16 + m) * 32 + 2 * hi;
      #pragma unroll
      for (int ni = 0; ni < 4; ++ni) {
        v8f acc = o[mi * 4 + ni];
        #pragma unroll
        for (int v = 0; v < 8; ++v) acc[v] *= arow[v];
        #pragma unroll
        for (int k2 = 0; k2 < 32; k2 += 4) {
          v2f a; a.x = pa[k2]; a.y = pa[k2 + 1];
          const float* vp = vbase + (size_t)(k2 + 2 * hi) * D_DIM + ni * 16;
          v2f bb; bb.x = vp[0]; bb.y = vp[D_DIM];
          acc = wmma_f32_k4(a, bb, acc);
        }
        o[mi * 4 + ni] = acc;
      }
    }
    __syncthreads();
  }

  // Epilogue: residual = x + O/l into Qs (reused)
  const float* xrow = x + ((size_t)b * S_LEN + q0) * D_DIM;
  #pragma unroll
  for (int mi = 0; mi < 4; ++mi) {
    #pragma unroll
    for (int ni = 0; ni < 4; ++ni) {
      const v8f acc = o[mi * 4 + ni];
      #pragma unroll
      for (int v = 0; v < 8; ++v) {
        const int r = mi * 16 + v + 8 * hi;
        const int c = ocol0 + ni * 16 + m;
        Qs[r * DPAD + c] = acc[v] / lstat[r] + xrow[(size_t)r * D_DIM + c];
      }
    }
  }
  __syncthreads();

  // LayerNorm stats: one thread per row
  if (tid < 64) {
    const float* rr = Qs + tid * DPAD;
    float su = 0.0f;
    for (int c = 0; c < D_DIM; ++c) su += rr[c];
    const float mu = su * (1.0f / D_DIM);
    float vs = 0.0f;
    for (int c = 0; c < D_DIM; ++c) { const float d = rr[c] - mu; vs += d * d; }
    mstat[tid] = mu;
    lstat[tid] = rsqrtf(vs * (1.0f / D_DIM) + 1e-6f);
  }
  __syncthreads();

  float* orow = out + ((size_t)b * S_LEN + q0) * D_DIM;
  for (int idx = tid; idx < 64 * D_DIM; idx += 256) {
    const int r = idx >> 9, c = idx & (D_DIM - 1);
    orow[(size_t)r * D_DIM + c] =
        (Qs[r * DPAD + c] - mstat[r]) * lstat[r] * gamma[c] + beta[c];
  }
}

// ---------------------------------------------------------------------------
extern "C" void kernel_launch(void* const* d_in, const int* in_sizes, int n_in,
                              void* d_out, int out_size, void* d_ws, size_t ws_size,
                              hipStream_t stream) {
  (void)in_sizes; (void)n_in; (void)out_size; (void)ws_size;
  const float* x     = (const float*)d_in[0];
  const int*   mask  = (const int*)  d_in[1];
  const float* Wq    = (const float*)d_in[2];
  const float* bq    = (const float*)d_in[3];
  const float* Wk    = (const float*)d_in[4];
  const float* bk    = (const float*)d_in[5];
  const float* Wv    = (const float*)d_in[6];
  const float* bv    = (const float*)d_in[7];
  const float* gamma = (const float*)d_in[8];
  const float* beta  = (const float*)d_in[9];
  float* outp = (float*)d_out;

  float* Qb = (float*)d_ws;
  float* Kb = Qb + (size_t)BATCH * S_LEN * D_DIM;
  float* Vb = Kb + (size_t)BATCH * S_LEN * D_DIM;

  qkv_proj_kernel<<<3072, 256, 0, stream>>>(x, Wq, bq, Wk, bk, Wv, bv, Qb, Kb, Vb);

  const size_t smem_bytes =
      (size_t)(64 * DPAD + 2 * KBUFF + 64 * 32 + 3 * 64) * sizeof(float); // ~273 KB
  flash_attn_ln_kernel<<<dim3(32, BATCH), 256, smem_bytes, stream>>>(
      Qb, Kb, Vb, x, mask, gamma, beta, outp);
}